// GraphConvolution_8856222564800
// MI455X (gfx1250) — compile-verified
//
#include <hip/hip_runtime.h>
#include <stdint.h>

// GraphConvolution: out = COO(rows,cols,vals) @ features,  E=1.6M, N=100K, D=32.
// Memory/atomic-bound scatter: lane==dim (wave32), async-LDS staged edge stream,
// agent-scope f32 atomics resolved in L2.
// NOTE: gc_coo_spmm is defined FIRST so the disasm snippet shows the hot loop.

#define DIM     32
#define CHUNK   128          // edges staged per wave per buffer (3 x b128 async loads)
#define WPB     8            // waves per block
#define BLOCK   (WPB * 32)

// ------------------------------------------------- async stage: 128 edges/wave
// Each lane pulls 16B (4 edges) of each array straight into this wave's LDS
// buffer via the CDNA5 async global->LDS path (tracked by ASYNCcnt).
__device__ __forceinline__ void gc_stage_async(const int* rows, const int* cols,
                                               const float* vals, int e0, int lane,
                                               uint32_t lds_r, uint32_t lds_c,
                                               uint32_t lds_v) {
  uint32_t goff = (uint32_t)(e0 + lane * 4) * 4u;   // byte offset into edge arrays
  uint32_t loff = (uint32_t)lane * 16u;             // byte offset into LDS buffer
  asm volatile("global_load_async_to_lds_b128 %0, %1, %2"
               :: "v"(lds_r + loff), "v"(goff), "s"(rows) : "memory");
  asm volatile("global_load_async_to_lds_b128 %0, %1, %2"
               :: "v"(lds_c + loff), "v"(goff), "s"(cols) : "memory");
  asm volatile("global_load_async_to_lds_b128 %0, %1, %2"
               :: "v"(lds_v + loff), "v"(goff), "s"(vals) : "memory");
}

__device__ __forceinline__ void gc_atomic_add(float* p, float v) {
  __hip_atomic_fetch_add(p, v, __ATOMIC_RELAXED, __HIP_MEMORY_SCOPE_AGENT);
}

// --------------------------------------------- consume one staged 128-edge chunk
__device__ __forceinline__ void gc_process_chunk(const float* __restrict__ feat,
                                                 float* __restrict__ out,
                                                 const int* shr, const int* shc,
                                                 const float* shv, int lane) {
#pragma unroll 4
  for (int j = 0; j < CHUNK; j += 4) {
    int4   r4 = *reinterpret_cast<const int4*>(shr + j);    // uniform LDS reads
    int4   c4 = *reinterpret_cast<const int4*>(shc + j);
    float4 v4 = *reinterpret_cast<const float4*>(shv + j);
    // 4 coalesced 128B feature gathers (L2-resident table) issued up front so
    // their latency overlaps; then 4 cacheline-wide f32 atomics.
    float g0 = __ldg(feat + (size_t)c4.x * DIM + lane);
    float g1 = __ldg(feat + (size_t)c4.y * DIM + lane);
    float g2 = __ldg(feat + (size_t)c4.z * DIM + lane);
    float g3 = __ldg(feat + (size_t)c4.w * DIM + lane);
    gc_atomic_add(out + (size_t)r4.x * DIM + lane, g0 * v4.x);
    gc_atomic_add(out + (size_t)r4.y * DIM + lane, g1 * v4.y);
    gc_atomic_add(out + (size_t)r4.z * DIM + lane, g2 * v4.z);
    gc_atomic_add(out + (size_t)r4.w * DIM + lane, g3 * v4.w);
  }
}

// ------------------------------------------------------------------ main kernel
__global__ __launch_bounds__(BLOCK) void gc_coo_spmm(
    const int* __restrict__ rows, const int* __restrict__ cols,
    const float* __restrict__ vals, const float* __restrict__ feat,
    float* __restrict__ out, int n_edges) {
  __shared__ __attribute__((aligned(16))) int   sh_rows[WPB][2][CHUNK];
  __shared__ __attribute__((aligned(16))) int   sh_cols[WPB][2][CHUNK];
  __shared__ __attribute__((aligned(16))) float sh_vals[WPB][2][CHUNK];

  const int lane   = threadIdx.x & 31;
  const int wave   = threadIdx.x >> 5;
  const int waveId = blockIdx.x * WPB + wave;
  const int nWaves = gridDim.x * WPB;
  const int nFull  = n_edges / CHUNK;

  // wave-relative LDS byte offsets of the two buffers of each array
  uint32_t lr[2], lc[2], lv[2];
  for (int b = 0; b < 2; ++b) {
    lr[b] = (uint32_t)(uintptr_t)(void*)&sh_rows[wave][b][0];
    lc[b] = (uint32_t)(uintptr_t)(void*)&sh_cols[wave][b][0];
    lv[b] = (uint32_t)(uintptr_t)(void*)&sh_vals[wave][b][0];
  }

  if (waveId < nFull) {
    int buf = 0;
    gc_stage_async(rows, cols, vals, waveId * CHUNK, lane, lr[0], lc[0], lv[0]);
    for (int c = waveId; c < nFull; c += nWaves) {
      int  cn   = c + nWaves;
      bool more = (cn < nFull);
      if (more)
        gc_stage_async(rows, cols, vals, cn * CHUNK, lane,
                       lr[buf ^ 1], lc[buf ^ 1], lv[buf ^ 1]);
      if (more)  // current chunk's 3 loads done (in-order), next 3 still in flight
        asm volatile("s_wait_asynccnt 0x3" ::: "memory");
      else
        asm volatile("s_wait_asynccnt 0x0" ::: "memory");
      gc_process_chunk(feat, out, &sh_rows[wave][buf][0], &sh_cols[wave][buf][0],
                       &sh_vals[wave][buf][0], lane);
      buf ^= 1;
    }
  }

  // tail (< CHUNK edges): one wave handles it straight from global (E%128==0
  // for this problem, but stay general).
  int rem = n_edges - nFull * CHUNK;
  if (rem > 0 && waveId == (nFull % nWaves)) {
    int e0 = nFull * CHUNK;
    for (int j = 0; j < rem; ++j) {
      int   r = __ldg(rows + e0 + j);
      int   c = __ldg(cols + e0 + j);
      float v = __ldg(vals + e0 + j);
      float g = __ldg(feat + (size_t)c * DIM + lane) * v;
      gc_atomic_add(out + (size_t)r * DIM + lane, g);
    }
  }
}

// ---------------------------------------------------------------- zero output
__global__ void gc_zero_out(float* __restrict__ p, int n) {
  int i = blockIdx.x * blockDim.x + threadIdx.x;
  int s = gridDim.x * blockDim.x;
  int n4 = n >> 2;
  float4 z; z.x = z.y = z.z = z.w = 0.0f;
  float4* __restrict__ p4 = reinterpret_cast<float4*>(p);
  for (int k = i; k < n4; k += s) p4[k] = z;
  for (int k = (n4 << 2) + i; k < n; k += s) p[k] = 0.0f;
}

// ------------------------------------------------------------------- launcher
extern "C" void kernel_launch(void* const* d_in, const int* in_sizes, int n_in,
                              void* d_out, int out_size, void* d_ws, size_t ws_size,
                              hipStream_t stream) {
  const int*   rows = (const int*)d_in[0];
  const int*   cols = (const int*)d_in[1];
  const float* vals = (const float*)d_in[2];
  const float* feat = (const float*)d_in[3];
  float*       out  = (float*)d_out;
  const int n_edges = in_sizes[0];

  // 1) zero the (poisoned) accumulator
  {
    int n4 = out_size >> 2;
    int blocks = (n4 + 255) / 256;
    if (blocks > 1024) blocks = 1024;
    if (blocks < 1)    blocks = 1;
    gc_zero_out<<<blocks, 256, 0, stream>>>(out, out_size);
  }

  // 2) pipelined scatter-add: ~2 chunks per wave so the double buffer is useful
  {
    int nFull  = n_edges / CHUNK;
    int blocks = (nFull + 2 * WPB - 1) / (2 * WPB);
    if (blocks < 1)    blocks = 1;
    if (blocks > 2048) blocks = 2048;
    gc_coo_spmm<<<blocks, BLOCK, 0, stream>>>(rows, cols, vals, feat, out, n_edges);
  }
}